// SimplePhotonicRaytracer_22643067585267
// MI455X (gfx1250) — compile-verified
//
#include <hip/hip_runtime.h>
#include <hip/hip_bf16.h>
#include <math.h>

// Problem geometry (from reference)
#define GG        81            // 9x9 cells
#define CB        16            // batches staged per chunk
#define NENT      810           // 10 digits x 81 cells
#define NENT_PAD  816           // pad to 51 tiles of 16 for WMMA
#define LSTR      5             // lut entry: r,g,b,mean,var

typedef __attribute__((ext_vector_type(2))) float v2f;
typedef __attribute__((ext_vector_type(4))) float v4f;
typedef __attribute__((ext_vector_type(8))) float v8f;
typedef __attribute__((ext_vector_type(4))) int   v4i;

#if defined(__AMDGCN__) && __has_builtin(__builtin_amdgcn_wmma_f32_16x16x4_f32)
#define USE_WMMA 1
#else
#define USE_WMMA 0
#endif

#if defined(__AMDGCN__) && __has_builtin(__builtin_amdgcn_global_load_async_to_lds_b128)
#define USE_ASYNC 1
#else
#define USE_ASYNC 0
#endif

// Single-wavelength optical response for (digit d, cell c, wavelength w)
__device__ __forceinline__ float resp_eval(int d, int c, int w,
                                           const float* __restrict__ ri,
                                           const float* __restrict__ ab,
                                           float scale) {
  const float lam0 = 6.5e-07f, lam1 = 5.5e-07f, lam2 = 4.5e-07f;
  float lam = (w == 0) ? lam0 : ((w == 1) ? lam1 : lam2);
  float n     = ri[c];
  float h     = 1e-4f + scale * (float)d;          // BASE_HEIGHT + ts*d
  float path  = h * n;
  float alpha = fabsf(ab[c * 3 + w]);
  float trans = __expf(-alpha * path);
  float opl   = 6.283185307179586f * path / lam;
  float interf = 0.5f * (1.0f + cosf(opl));        // full-accuracy cos (arg ~2e3 rad)
  float r1    = (1.0f - n) / (1.0f + n);
  float tf    = 1.0f - r1 * r1;                    // 1 - Fresnel R
  return tf * trans * interf;
}

__global__ __launch_bounds__(256) void photonic_kernel(
    const int* __restrict__ grid, const float* __restrict__ ri,
    const float* __restrict__ ab, const float* __restrict__ ts,
    float* __restrict__ feats, float* __restrict__ resp, int nchunk) {
  __shared__ __align__(16) float lut[NENT_PAD * LSTR];   // 16320 B
  __shared__ __align__(16) int   stage[CB * GG];         //  5184 B

  const int tid = threadIdx.x;
  const float scale = ts[0];

  // ---------------- Phase 0: build (d,c) LUT in LDS ----------------
#if USE_WMMA
  {
    const int lane  = tid & 31;
    const int wave  = tid >> 5;
    const int nwave = blockDim.x >> 5;
    const int hi    = lane & 16;   // 0 => lanes 0-15 (K0,K1); 16 => lanes 16-31 (K2,K3)
    const int m     = lane & 15;   // M row = entry within tile
    // B (4x16): rows K0..K2 = 1/3, row K3 = 0.  VGPR0 = 1/3 everywhere,
    // VGPR1 = 1/3 (lo half) / 0 (hi half) — correct under either half ordering.
    v2f bm; bm.x = 1.0f / 3.0f; bm.y = hi ? 0.0f : (1.0f / 3.0f);
    for (int tile = wave; tile < NENT_PAD / 16; tile += nwave) {  // wave-uniform: EXEC all 1s
      int e = tile * 16 + m;
      int d = e / GG;              // pad entries get d==10 -> harmless, never read
      int c = e - d * GG;
      float v0 = resp_eval(d, c, hi ? 2 : 0, ri, ab, scale);      // K0=r | K2=b
      float v1 = hi ? 0.0f : resp_eval(d, c, 1, ri, ab, scale);   // K1=g | K3=0
      v2f a;  a.x  = v0;      a.y  = v1;
      v2f a2; a2.x = v0 * v0; a2.y = v1 * v1;
      v8f cz = {};
      v8f dm = __builtin_amdgcn_wmma_f32_16x16x4_f32(false, a,  false, bm, (short)0, cz, false, false);
      v8f dq = __builtin_amdgcn_wmma_f32_16x16x4_f32(false, a2, false, bm, (short)0, cz, false, false);
      if (!hi) { lut[e * LSTR + 0] = v0; lut[e * LSTR + 1] = v1; }
      else     { lut[e * LSTR + 2] = v0; }
      if (m == 0) {  // D layout: VGPR v = row M=v (lanes 0-15) / M=v+8 (lanes 16-31)
        int base = tile * 16 + (hi ? 8 : 0);
#pragma unroll
        for (int v = 0; v < 8; ++v) {
          float mu = dm[v], ms = dq[v];
          lut[(base + v) * LSTR + 3] = mu;
          lut[(base + v) * LSTR + 4] = 1.5f * (ms - mu * mu);  // ddof=1, 3 samples
        }
      }
    }
  }
#else
  for (int e = tid; e < NENT_PAD; e += blockDim.x) {
    int d = e / GG, c = e - d * GG;
    float r = resp_eval(d, c, 0, ri, ab, scale);
    float g = resp_eval(d, c, 1, ri, ab, scale);
    float b = resp_eval(d, c, 2, ri, ab, scale);
    float mu = (r + g + b) * (1.0f / 3.0f);
    float var = ((r - mu) * (r - mu) + (g - mu) * (g - mu) + (b - mu) * (b - mu)) * 0.5f;
    lut[e * LSTR + 0] = r; lut[e * LSTR + 1] = g; lut[e * LSTR + 2] = b;
    lut[e * LSTR + 3] = mu; lut[e * LSTR + 4] = var;
  }
#endif
  __syncthreads();

  // ---------------- Phase 1: stream batches through LDS ----------------
  for (int chunk = (int)blockIdx.x; chunk < nchunk; chunk += (int)gridDim.x) {
    const int base = chunk * (CB * GG);  // element offset, 16B-aligned (5184 B stride)
#if USE_ASYNC
    for (int i = tid; i < (CB * GG) / 4; i += blockDim.x) {
      __builtin_amdgcn_global_load_async_to_lds_b128(
          (__attribute__((address_space(1))) v4i*)(grid + base + i * 4),
          (__attribute__((address_space(3))) v4i*)(&stage[i * 4]), 0, 0);
    }
#if __has_builtin(__builtin_amdgcn_s_wait_asynccnt)
    __builtin_amdgcn_s_wait_asynccnt(0);
#else
    asm volatile("s_wait_asynccnt 0" ::: "memory");
#endif
#else
    for (int i = tid; i < CB * GG; i += blockDim.x) stage[i] = grid[base + i];
#endif
    __syncthreads();

    for (int i = tid; i < CB * GG; i += blockDim.x) {
      int bl  = i / GG;
      int c   = i - bl * GG;
      int row = c / 9;
      int col = c - row * 9;
      int d   = stage[i];
      const float* L = &lut[(d * GG + c) * LSTR];
      float r = L[0], g = L[1], b = L[2], mu = L[3], var = L[4];
      float gx = 0.0f, gy = 0.0f;                     // diff append => 0 at edges
      if (col < 8) gx = lut[(stage[i + 1] * GG + c + 1) * LSTR + 3] - mu;
      if (row < 8) gy = lut[(stage[i + 9] * GG + c + 9) * LSTR + 3] - mu;
      int idx = base + i;                             // global b*81 + c (< 2^24)
      v4f f; f.x = mu; f.y = var; f.z = gx; f.w = gy;
      __builtin_nontemporal_store(f, (v4f*)(feats + (size_t)idx * 4));
      __builtin_nontemporal_store(r, resp + (size_t)idx * 3 + 0);
      __builtin_nontemporal_store(g, resp + (size_t)idx * 3 + 1);
      __builtin_nontemporal_store(b, resp + (size_t)idx * 3 + 2);
    }
    __syncthreads();  // stage reuse safety before next chunk's async writes
  }
}

extern "C" void kernel_launch(void* const* d_in, const int* in_sizes, int n_in,
                              void* d_out, int out_size, void* d_ws, size_t ws_size,
                              hipStream_t stream) {
  const int*   grid = (const int*)d_in[0];
  const float* ri   = (const float*)d_in[1];
  const float* ab   = (const float*)d_in[2];
  const float* ts   = (const float*)d_in[3];
  const int nbatch  = in_sizes[0] / GG;      // 131072
  const int nchunk  = nbatch / CB;           // 8192
  float* feats = (float*)d_out;
  float* resp  = feats + (size_t)nbatch * GG * 4;
  dim3 block(256);
  dim3 g(2048);                              // grid-stride, 8 wave32 per block
  hipLaunchKernelGGL(photonic_kernel, g, block, 0, stream,
                     grid, ri, ab, ts, feats, resp, nchunk);
}